// SpectralConv2d_60971355734795
// MI455X (gfx1250) — compile-verified
//
#include <hip/hip_runtime.h>
#include <math.h>

// Problem constants (match reference)
#define BATCH 8
#define NPTS  4096
#define NCIN  32
#define NCOUT 32
#define MM1   16
#define MM2   16
// forward: 512 complex modes (x 0..31, y 0..15) -> 1024 real columns
// inverse: 992 complex modes (x 0..31, y 0..30) -> K = 1984

typedef float v2f __attribute__((ext_vector_type(2)));
typedef float v8f __attribute__((ext_vector_type(8)));

#define TWO_PI 6.28318530717958647692f

// Workspace layout (floats):
//   emb      : [B][1024]                      @ 0       (8192)
//   u_ft     : [B][32][1024]                  @ 8192    (262144)
//   out_full : [B][32][32][31][2]             @ 270336  (507904)
#define WS_EMB   0
#define WS_UFT   8192
#define WS_OFULL 270336

__device__ __forceinline__ v8f wmma_f32x4(v2f a, v2f b, v8f c) {
  // D = A(16x4 f32) * B(4x16 f32) + C(16x16 f32)
  return __builtin_amdgcn_wmma_f32_16x16x4_f32(false, a, false, b, (short)0, c,
                                               false, false);
}

// ---------------------------------------------------------------------------
// Kernel 1: conditioning  h = t @ w_freq + b_freq   -> emb ws [B][1024]
// ---------------------------------------------------------------------------
__global__ __launch_bounds__(256) void k1_emb(const float* __restrict__ t,
                                              const float* __restrict__ wf,
                                              const float* __restrict__ bf,
                                              float* __restrict__ emb) {
  __shared__ float st[128];
  const int b = blockIdx.x;
  if (threadIdx.x < 128) st[threadIdx.x] = t[b * 128 + threadIdx.x];
  __syncthreads();
  for (int j = threadIdx.x; j < 1024; j += 256) {
    float acc = bf[j];
    for (int k = 0; k < 128; ++k) acc = fmaf(st[k], wf[k * 1024 + j], acc);
    emb[b * 1024 + j] = acc;
  }
}

// ---------------------------------------------------------------------------
// Kernel 2a: forward NUFFT GEMM with fp32 WMMA.
//   u_ft[b][c][j] ,  j = (x*16+y)*2 + part,  x in 0..31, y in 0..15
// grid (8 col-groups of 128, 8 batches), 512 threads = 16 waves.
// ---------------------------------------------------------------------------
__global__ __launch_bounds__(512) void k2a_fwd(const float* __restrict__ u,
                                               const float* __restrict__ xin,
                                               float* __restrict__ uft) {
  extern __shared__ float smem[];
  float* sU  = smem;         // [32][64]      2048
  float* sE1 = smem + 2048;  // [64][32][2]   4096  (exp(-2pi i x0 k1[x]))
  float* sE2 = smem + 6144;  // [64][16][2]   2048  (exp(-2pi i x1 y))
  const int b = blockIdx.y;
  const int tid = (int)threadIdx.x;
  const int lane = tid & 31, wave = tid >> 5;
  const int mtile = wave >> 3, nt = wave & 7;
  const int lm = lane & 15, khalf = lane >> 4;
  const int j = (int)blockIdx.x * 128 + nt * 16 + lm;  // output column
  const int mode = j >> 1, part = j & 1;
  const int mx = mode >> 4, my = mode & 15;

  v8f acc = {};

  for (int kc = 0; kc < 64; ++kc) {  // 64 chunks of 64 points over K=4096
    // stage u tile [32][64]
    for (int idx = tid; idx < 2048; idx += 512) {
      int c = idx >> 6, kk = idx & 63;
      sU[idx] = u[(b * NCIN + c) * NPTS + kc * 64 + kk];
    }
    // stage per-point basis factor tables (rank-1 NUFFT basis)
    for (int idx = tid; idx < 64 * 48; idx += 512) {
      int nn = idx / 48;
      int col = idx - nn * 48;
      int n = kc * 64 + nn;
      if (col < 32) {
        float k1 = (float)(col < 16 ? col : col - 32);
        float ang = -TWO_PI * xin[(b * NPTS + n) * 2 + 0] * k1;
        float s, c2;
        __sincosf(ang, &s, &c2);
        sE1[(nn * 32 + col) * 2 + 0] = c2;
        sE1[(nn * 32 + col) * 2 + 1] = s;
      } else {
        int y = col - 32;  // 0..15, k2[y] = y
        float ang = -TWO_PI * xin[(b * NPTS + n) * 2 + 1] * (float)y;
        float s, c2;
        __sincosf(ang, &s, &c2);
        sE2[(nn * 16 + y) * 2 + 0] = c2;
        sE2[(nn * 16 + y) * 2 + 1] = s;
      }
    }
    __syncthreads();

#pragma unroll
    for (int ks = 0; ks < 16; ++ks) {  // 16 steps of K=4
      v2f a = *(const v2f*)&sU[(mtile * 16 + lm) * 64 + ks * 4 + 2 * khalf];
      v2f bfrag;
#pragma unroll
      for (int tE = 0; tE < 2; ++tE) {
        int nn = ks * 4 + 2 * khalf + tE;  // point within chunk
        v2f e1 = *(const v2f*)&sE1[(nn * 32 + mx) * 2];
        v2f e2 = *(const v2f*)&sE2[(nn * 16 + my) * 2];
        float br = e1.x * e2.x - e1.y * e2.y;
        float bi = e1.x * e2.y + e1.y * e2.x;
        bfrag[tE] = part ? bi : br;  // cndmask, EXEC stays all-ones
      }
      acc = wmma_f32x4(a, bfrag, acc);
    }
    __syncthreads();
  }

#pragma unroll
  for (int r = 0; r < 8; ++r) {
    int mrow = mtile * 16 + khalf * 8 + r;
    uft[(b * NCIN + mrow) * 1024 + j] = acc[r];
  }
}

// ---------------------------------------------------------------------------
// Kernel 2b: emb conditioning + complex channel mixing + Hermitian extension.
//   out_full[b][o][x][yy][ri], yy 0..30 (row stride 62 floats)
// ---------------------------------------------------------------------------
__global__ __launch_bounds__(256) void k2b_mix(const float* __restrict__ uft,
                                               const float* __restrict__ emb,
                                               const float* __restrict__ w1,
                                               const float* __restrict__ w2,
                                               float* __restrict__ ofull) {
  const int b = blockIdx.x;
  for (int task = threadIdx.x; task < NCOUT * 512; task += 256) {
    int o = task >> 9;
    int mode = task & 511;
    int x = mode >> 4, y = mode & 15;
    int x16 = x & 15;
    int which = x >> 4;  // 0 -> emb1/w1 (x 0..15), 1 -> emb2/w2 (x 16..31)
    const float* w = which ? w2 : w1;
    int eb = b * 1024 + x16 * 64 + y * 4 + which * 2;  // h[b,x16,y,which,:]
    float er = emb[eb], ei = emb[eb + 1];
    float accr = 0.f, acci = 0.f;
    for (int i = 0; i < NCIN; ++i) {
      int ub = (b * NCIN + i) * 1024 + mode * 2;
      float ur = uft[ub], ui = uft[ub + 1];
      float ar = ur * er - ui * ei;  // u_ft * emb
      float ai = ur * ei + ui * er;
      int wb = i * 16384 + o * 512 + x16 * 32 + y * 2;  // w[i,o,x16,y,:]
      float wr = w[wb], wi = w[wb + 1];
      accr += ar * wr - ai * wi;
      acci += ar * wi + ai * wr;
    }
    int fb = ((b * NCOUT + o) * 32 + x) * 62;
    ofull[fb + y * 2 + 0] = accr;
    ofull[fb + y * 2 + 1] = acci;
    if (y >= 1) {  // Hermitian: out_full[31-x, 31-y] = conj(out_ft[x, y])
      int fb2 = ((b * NCOUT + o) * 32 + (31 - x)) * 62;
      ofull[fb2 + (31 - y) * 2 + 0] = accr;
      ofull[fb2 + (31 - y) * 2 + 1] = -acci;
    }
  }
}

// ---------------------------------------------------------------------------
// Kernel 3: inverse NUFFT.  Y[b][c][n] = sum_m Fr*gr - Fi*gi  (K = 1984)
// grid (32 n-tiles of 128, 8 batches), 512 threads = 16 waves, 1 tile/wave.
// ---------------------------------------------------------------------------
__global__ __launch_bounds__(512) void k3_inv(const float* __restrict__ xout,
                                              const float* __restrict__ ofull,
                                              float* __restrict__ Y) {
  extern __shared__ float smem[];
  float* sE1 = smem;          // [128][32][2]  8192  (exp(+2pi i x0 k1[x]))
  float* sE2 = smem + 8192;   // [128][31][2]  7936  (exp(+2pi i x1 k2[y]))
  float* sA  = smem + 16128;  // [32][64]      2048  A chunk (Fr, -Fi)
  const int b = blockIdx.y;
  const int n0 = (int)blockIdx.x * 128;
  const int tid = (int)threadIdx.x;
  const int lane = tid & 31, wave = tid >> 5;
  const int mtile = wave >> 3, nsub = wave & 7;
  const int lm = lane & 15, khalf = lane >> 4;
  const int nn = nsub * 16 + lm;  // point within tile owned by this lane col

  // per-point basis factor tables, computed once per block
  for (int idx = tid; idx < 128 * 63; idx += 512) {
    int pn = idx / 63;
    int col = idx - pn * 63;
    int n = n0 + pn;
    if (col < 32) {
      float k = (float)(col < 16 ? col : col - 32);
      float ang = TWO_PI * xout[(b * NPTS + n) * 2 + 0] * k;
      float s, c;
      __sincosf(ang, &s, &c);
      sE1[(pn * 32 + col) * 2 + 0] = c;
      sE1[(pn * 32 + col) * 2 + 1] = s;
    } else {
      int y = col - 32;  // 0..30
      float k = (float)(y < 16 ? y : y - 31);
      float ang = TWO_PI * xout[(b * NPTS + n) * 2 + 1] * k;
      float s, c;
      __sincosf(ang, &s, &c);
      sE2[(pn * 31 + y) * 2 + 0] = c;
      sE2[(pn * 31 + y) * 2 + 1] = s;
    }
  }
  __syncthreads();

  v8f acc = {};
  for (int mc = 0; mc < 31; ++mc) {  // 31 chunks x 32 modes = 992 modes
    // stage A chunk: K index kk = 2*(m - mc*32) + part ; odd K holds -Fi
    for (int idx = tid; idx < 2048; idx += 512) {
      int c = idx >> 6, kk = idx & 63;
      int m = mc * 32 + (kk >> 1);
      int x = m / 31, y = m - x * 31;
      int prt = kk & 1;
      float v = ofull[((b * NCOUT + c) * 32 + x) * 62 + y * 2 + prt];
      sA[idx] = prt ? -v : v;
    }
    __syncthreads();

#pragma unroll
    for (int ks = 0; ks < 16; ++ks) {
      v2f a = *(const v2f*)&sA[(mtile * 16 + lm) * 64 + ks * 4 + 2 * khalf];
      v2f bfrag;
#pragma unroll
      for (int tE = 0; tE < 2; ++tE) {
        int kkk = ks * 4 + 2 * khalf + tE;
        int m = mc * 32 + (kkk >> 1);
        int prt = kkk & 1;
        int x = m / 31, y = m - x * 31;
        v2f e1 = *(const v2f*)&sE1[(nn * 32 + x) * 2];
        v2f e2 = *(const v2f*)&sE2[(nn * 31 + y) * 2];
        float br = e1.x * e2.x - e1.y * e2.y;
        float bi = e1.x * e2.y + e1.y * e2.x;
        bfrag[tE] = prt ? bi : br;
      }
      acc = wmma_f32x4(a, bfrag, acc);
    }
    __syncthreads();
  }

#pragma unroll
  for (int r = 0; r < 8; ++r) {
    int c = mtile * 16 + khalf * 8 + r;
    int n = n0 + nsub * 16 + lm;
    Y[(b * NCOUT + c) * NPTS + n] = acc[r];
  }
}

// ---------------------------------------------------------------------------
extern "C" void kernel_launch(void* const* d_in, const int* in_sizes, int n_in,
                              void* d_out, int out_size, void* d_ws,
                              size_t ws_size, hipStream_t stream) {
  const float* u      = (const float*)d_in[0];  // [8,32,4096]
  const float* x_in   = (const float*)d_in[1];  // [8,4096,2]
  const float* x_out  = (const float*)d_in[2];  // [8,4096,2]
  const float* t      = (const float*)d_in[3];  // [8,128]
  const float* w_freq = (const float*)d_in[4];  // [128,1024]
  const float* b_freq = (const float*)d_in[5];  // [1,1024]
  const float* w1     = (const float*)d_in[6];  // [32,32,16,16,2]
  const float* w2     = (const float*)d_in[7];  // [32,32,16,16,2]
  float* out = (float*)d_out;                   // [8,32,4096]
  float* ws = (float*)d_ws;

  float* emb   = ws + WS_EMB;
  float* uft   = ws + WS_UFT;
  float* ofull = ws + WS_OFULL;

  k1_emb<<<dim3(BATCH), dim3(256), 0, stream>>>(t, w_freq, b_freq, emb);

  // forward NUFFT: LDS = 2048 + 4096 + 2048 floats = 32 KB
  k2a_fwd<<<dim3(8, BATCH), dim3(512), 8192 * sizeof(float), stream>>>(u, x_in,
                                                                       uft);

  k2b_mix<<<dim3(BATCH), dim3(256), 0, stream>>>(uft, emb, w1, w2, ofull);

  // inverse NUFFT: LDS = 8192 + 7936 + 2048 floats = 71 KB
  k3_inv<<<dim3(32, BATCH), dim3(512), 18176 * sizeof(float), stream>>>(
      x_out, ofull, out);
}